// GWVAE_67997922230796
// MI455X (gfx1250) — compile-verified
//
#include <hip/hip_runtime.h>
#include <hip/hip_bf16.h>

#define Bg 32
#define Nn 1024
#define Ee 32768
#define Ff 32
#define Cc 10
#define ZS 1040   // K padded to multiple of 4 (1034 -> 1040, zeros beyond 1034)
#define KW 1034   // true K = N + C

typedef float v2f __attribute__((ext_vector_type(2)));
typedef float v8f __attribute__((ext_vector_type(8)));

__device__ __forceinline__ float fsig(float v) {
    // sigmoid with v_rcp_f32 instead of the IEEE division refinement chain
    return __builtin_amdgcn_rcpf(1.0f + __expf(-v));
}

// xs[b,n] = sum_f x[b,n,f]
__global__ void k_xsum(const float* __restrict__ x, float* __restrict__ xs) {
    int t = blockIdx.x * blockDim.x + threadIdx.x;   // B*N
    const float4* g = (const float4*)(x + (long)t * Ff);
    float s = 0.f;
#pragma unroll
    for (int i = 0; i < 8; ++i) {
        float4 v = g[i];
        s += v.x + v.y + v.z + v.w;
    }
    xs[t] = s;
}

__global__ void k_deg_init(float* __restrict__ deg) {
    int t = blockIdx.x * blockDim.x + threadIdx.x;   // B*N
    deg[t] = 1.0f;                                   // self loop
}

__global__ void k_deg_atomic(const int* __restrict__ edges, float* __restrict__ deg) {
    int t = blockIdx.x * blockDim.x + threadIdx.x;   // B*E
    int b = t >> 15;                                 // E = 2^15
    int e = t & (Ee - 1);
    int row = edges[b * 2 * Ee + e];
    atomicAdd(&deg[b * Nn + row], 1.0f);
}

// dis = rsqrt(deg); tt = self-loop term = xs/deg
__global__ void k_t_init(const float* __restrict__ deg, const float* __restrict__ xs,
                         float* __restrict__ dis, float* __restrict__ tt) {
    int t = blockIdx.x * blockDim.x + threadIdx.x;   // B*N
    float d = deg[t];
    dis[t] = rsqrtf(d);
    tt[t] = xs[t] / d;
}

// scalar edge scatter: tt[b,row] += dis[row]*dis[col]*xs[b,col]   (1 atomic / edge)
__global__ void k_spmm_edge(const int* __restrict__ edges, const float* __restrict__ dis,
                            const float* __restrict__ xs, float* __restrict__ tt) {
    int t = blockIdx.x * blockDim.x + threadIdx.x;   // B*E
    int b = t >> 15;
    int e = t & (Ee - 1);
    int row = edges[b * 2 * Ee + e];
    int col = edges[b * 2 * Ee + Ee + e];
    float w = dis[b * Nn + row] * dis[b * Nn + col] * xs[b * Nn + col];
    atomicAdd(&tt[b * Nn + row], w);
}

// S[b,n] = phi[b,n,:] . tt[b,:]   (one wave per output element; streams phi once)
__global__ void k_matvec(const float* __restrict__ phi, const float* __restrict__ tt,
                         float* __restrict__ S) {
    int wave = (blockIdx.x * blockDim.x + threadIdx.x) >> 5;   // B*N waves
    int lane = threadIdx.x & 31;
    int b = wave >> 10;
    const float* prow = phi + (long)wave * Nn;
    const float* tb = tt + b * Nn;
    float sum = 0.f;
#pragma unroll
    for (int i = 0; i < 8; ++i) {
        int k = (i * 32 + lane) * 4;
        __builtin_prefetch(prow + Nn + k, 0, 0);     // next row (global_prefetch_b8)
        float4 p = *(const float4*)(prow + k);
        float4 tv = *(const float4*)(tb + k);
        sum += p.x * tv.x + p.y * tv.y + p.z * tv.z + p.w * tv.w;
    }
#pragma unroll
    for (int off = 16; off; off >>= 1) sum += __shfl_xor(sum, off, 32);
    if (lane == 0) S[wave] = sum;
}

// heads: x_mu/x_logvar/a_mu/a_logvar outputs + reparam + padded decoder inputs Z, Za
__global__ void k_heads(const float* __restrict__ S,
                        const float* __restrict__ fx1, const float* __restrict__ fx2,
                        const float* __restrict__ fa1, const float* __restrict__ fa2,
                        const float* __restrict__ eps_x, const float* __restrict__ eps_a,
                        const float* __restrict__ ys,
                        float* __restrict__ o_xmu, float* __restrict__ o_xlv,
                        float* __restrict__ o_amu, float* __restrict__ o_alv,
                        float* __restrict__ Z, float* __restrict__ Za) {
    int t = blockIdx.x * blockDim.x + threadIdx.x;   // B*N
    int b = t >> 10;
    int n = t & (Nn - 1);
    float s = S[t];
    float xm = fx1[n] * s, xlv = fx2[n] * s;
    float am = fa1[n] * s, alv = fa2[n] * s;
    o_xmu[t] = xm; o_xlv[t] = xlv; o_amu[t] = am; o_alv[t] = alv;
    float xz = xm + eps_x[t] * __expf(0.5f * xlv);
    float az = am + eps_a[t] * __expf(0.5f * alv);
    Z[b * ZS + n] = xz;
    Za[b * ZS + n] = az;
    if (n < (ZS - Nn)) {                             // pad region [1024,1040): ys then zeros
        float v = (n < Cc) ? ys[b * Cc + n] : 0.0f;
        Z[b * ZS + Nn + n] = v;
        Za[b * ZS + Nn + n] = v;
    }
}

// GEMM: out[32, Nout] = act( Z[32,ZS(pad)] * W[Nout,KW]^T + bias ) via V_WMMA_F32_16X16X4_F32
// One wave per 16x16 output tile. A lane layout: lanes0-15 M=lane (K=k0,k0+1),
// lanes16-31 M=lane-16 (K=k0+2,k0+3). B mirrored with N in lane%16.
__global__ void k_gemm_wmma(const float* __restrict__ Z, const float* __restrict__ W,
                            const float* __restrict__ bias, float* __restrict__ out,
                            int Nout, int applySig) {
    int wave = (blockIdx.x * blockDim.x + threadIdx.x) >> 5;
    int lane = threadIdx.x & 31;
    int ntiles = Nout >> 4;
    int mt = wave / ntiles;                          // 0..1  (B=32 -> 2 M-tiles)
    int nt = wave - mt * ntiles;
    int laneHi = lane >> 4;
    int lane16 = lane & 15;
    const float* zrow = Z + (mt * 16 + lane16) * ZS;
    const float* wrow = W + (long)(nt * 16 + lane16) * KW;
    int kb = 2 * laneHi;
    v8f c = {0.f, 0.f, 0.f, 0.f, 0.f, 0.f, 0.f, 0.f};
    for (int k0 = 0; k0 < 1032; k0 += 4) {
        int kA = k0 + kb;
        v2f a; a.x = zrow[kA];  a.y = zrow[kA + 1];
        v2f bm; bm.x = wrow[kA]; bm.y = wrow[kA + 1];
        c = __builtin_amdgcn_wmma_f32_16x16x4_f32(false, a, false, bm,
                                                  (short)0, c, false, false);
    }
    {   // tail: k = 1032..1035 ; Z zero-padded at 1034/1035, W guarded (KW=1034)
        int kA = 1032 + kb;
        v2f a; a.x = zrow[kA]; a.y = zrow[kA + 1];
        v2f bm;
        if (laneHi == 0) { bm.x = wrow[1032]; bm.y = wrow[1033]; }
        else             { bm.x = 0.f;        bm.y = 0.f; }
        c = __builtin_amdgcn_wmma_f32_16x16x4_f32(false, a, false, bm,
                                                  (short)0, c, false, false);
    }
    int n = nt * 16 + lane16;
    float bv = bias[n];
#pragma unroll
    for (int v = 0; v < 8; ++v) {
        int m = v + 8 * laneHi;                      // C/D layout: VGPR v, lane group
        float val = c[v] + bv;
        if (applySig) val = fsig(val);
        out[(long)(mt * 16 + m) * Nout + n] = val;
    }
}

// recon_a = sigmoid(za_i * za_j), float4 stores
__global__ void k_recon_a(const float* __restrict__ za, float* __restrict__ out) {
    long t = (long)blockIdx.x * blockDim.x + threadIdx.x;  // B*N*N/4
    int j0 = (int)(t & 255) << 2;                          // N/4 = 256 groups
    long bi = t >> 8;                                      // b*N + i
    int b = (int)(bi >> 10);
    float zi = za[bi];
    float4 zj = *(const float4*)(za + b * Nn + j0);
    float4 r;
    r.x = fsig(zi * zj.x);
    r.y = fsig(zi * zj.y);
    r.z = fsig(zi * zj.z);
    r.w = fsig(zi * zj.w);
    *(float4*)(out + bi * Nn + j0) = r;
}

extern "C" void kernel_launch(void* const* d_in, const int* in_sizes, int n_in,
                              void* d_out, int out_size, void* d_ws, size_t ws_size,
                              hipStream_t stream) {
    const float* x     = (const float*)d_in[0];
    const float* phi   = (const float*)d_in[1];
    const float* ys    = (const float*)d_in[2];
    const float* fx1   = (const float*)d_in[3];
    const float* fx2   = (const float*)d_in[4];
    const float* fa1   = (const float*)d_in[5];
    const float* fa2   = (const float*)d_in[6];
    const float* Wx    = (const float*)d_in[7];
    const float* bx    = (const float*)d_in[8];
    const float* Wa    = (const float*)d_in[9];
    const float* ba    = (const float*)d_in[10];
    const float* eps_x = (const float*)d_in[11];
    const float* eps_a = (const float*)d_in[12];
    const int*   edges = (const int*)d_in[13];

    float* out = (float*)d_out;
    float* o_rx  = out;                       // recon_x  [B,N,F]
    float* o_xmu = out + 1048576;             // x_mu     [B,N]
    float* o_xlv = out + 1048576 + 32768;     // x_logvar [B,N]
    float* o_ra  = out + 1114112;             // recon_a  [B,N,N]
    float* o_amu = out + 34668544;            // a_mu     [B,N]
    float* o_alv = out + 34701312;            // a_logvar [B,N]

    float* ws  = (float*)d_ws;
    float* deg = ws;                          // B*N
    float* dis = deg + Bg * Nn;               // B*N
    float* xs  = dis + Bg * Nn;               // B*N
    float* tt  = xs + Bg * Nn;                // B*N
    float* S   = tt + Bg * Nn;                // B*N
    float* Z   = S + Bg * Nn;                 // B*ZS
    float* Za  = Z + Bg * ZS;                 // B*ZS
    float* za  = Za + Bg * ZS;                // B*N

    const int T = 256;
    k_xsum      <<<(Bg * Nn) / T, T, 0, stream>>>(x, xs);
    k_deg_init  <<<(Bg * Nn) / T, T, 0, stream>>>(deg);
    k_deg_atomic<<<(Bg * Ee) / T, T, 0, stream>>>(edges, deg);
    k_t_init    <<<(Bg * Nn) / T, T, 0, stream>>>(deg, xs, dis, tt);
    k_spmm_edge <<<(Bg * Ee) / T, T, 0, stream>>>(edges, dis, xs, tt);
    k_matvec    <<<(Bg * Nn * 32) / T, T, 0, stream>>>(phi, tt, S);
    k_heads     <<<(Bg * Nn) / T, T, 0, stream>>>(S, fx1, fx2, fa1, fa2,
                                                  eps_x, eps_a, ys,
                                                  o_xmu, o_xlv, o_amu, o_alv, Z, Za);
    // decoder_x: 2 x 2048 tiles -> 4096 waves -> 512 blocks of 8 waves
    k_gemm_wmma <<<512, T, 0, stream>>>(Z, Wx, bx, o_rx, Nn * Ff, 1);
    // decoder_a: 2 x 64 tiles -> 128 waves -> 16 blocks
    k_gemm_wmma <<<16, T, 0, stream>>>(Za, Wa, ba, za, Nn, 0);
    k_recon_a   <<<(int)(((long)Bg * Nn * Nn / 4) / T), T, 0, stream>>>(za, o_ra);
}